// CorrelationLayer_8546984919481
// MI455X (gfx1250) — compile-verified
//
#include <hip/hip_runtime.h>

typedef __attribute__((ext_vector_type(2))) float v2f;
typedef __attribute__((ext_vector_type(8))) float v8f;

#define B_TOT            4096
#define C_CH             512
#define HW               16        // 4x4 spatial, flattened
#define WAVES_PER_BLOCK  8

__global__ __launch_bounds__(32 * WAVES_PER_BLOCK)
void corr_wmma_f32_kernel(const float* __restrict__ feat1,
                          const float* __restrict__ feat2,
                          float* __restrict__ out) {
    __shared__ float red[WAVES_PER_BLOCK * 16];

    const int tid  = threadIdx.x;
    const int wave = tid >> 5;          // wave32
    const int lane = tid & 31;
    const int b    = blockIdx.x * WAVES_PER_BLOCK + wave;   // one batch per wave

    const size_t base = (size_t)b * (C_CH * HW);
    const float* __restrict__ Ap = feat1 + base;   // A[M=i, K=c]  = feat1[b,c,i] @ c*16+i
    const float* __restrict__ Bp = feat2 + base;   // B[K=c, N=j]  = feat2[b,c,j] @ c*16+j

    // WMMA f32 16x16x4 fragment addressing (wave32):
    //  A: lane<16 holds M=lane, K = {0,1};  lane>=16 holds M=lane-16, K = {2,3}
    //  B: lane<16 holds N=lane, K = {0,1};  lane>=16 holds N=lane-16, K = {2,3}
    const int i    = lane & 15;      // M index for A / N index for B
    const int kSel = lane >> 4;      // which K-half this lane carries
    const int off  = kSel * 2 * HW + i;

    v8f acc = {0.f, 0.f, 0.f, 0.f, 0.f, 0.f, 0.f, 0.f};

    #pragma unroll 8
    for (int c0 = 0; c0 < C_CH; c0 += 4) {
        v2f a, bf;
        a.x  = __builtin_nontemporal_load(&Ap[c0 * HW + off]);       // K = c0+2*kSel
        a.y  = __builtin_nontemporal_load(&Ap[c0 * HW + off + HW]);  // K = c0+2*kSel+1
        bf.x = __builtin_nontemporal_load(&Bp[c0 * HW + off]);
        bf.y = __builtin_nontemporal_load(&Bp[c0 * HW + off + HW]);
        // G += A(16x4) x B(4x16), exact fp32
        acc = __builtin_amdgcn_wmma_f32_16x16x4_f32(false, a, false, bf,
                                                    (short)0, acc, false, false);
    }

    // --- epilogue: scatter-reduce G[(p,q),(u,v)] into 16 displacement outputs ---
    if (lane < 16) red[wave * 16 + lane] = 0.f;
    __syncthreads();

    const int u = i >> 2;            // feat2 spatial row (N = u*4+v)
    const int v = i & 3;             // feat2 spatial col
    const int mbase = kSel * 8;      // C/D layout: VGPR r holds M = r (+8 for upper lanes)
    #pragma unroll
    for (int r = 0; r < 8; ++r) {
        const int m = mbase + r;     // feat1 spatial index (M = p*4+q)
        const int p = m >> 2;
        const int q = m & 3;
        const int y = p - u + 2;     // displacement row
        const int x = q - v + 2;     // displacement col
        if ((unsigned)y < 4u && (unsigned)x < 4u)
            atomicAdd(&red[wave * 16 + (y * 4 + x)], acc[r]);   // ds_add_f32
    }
    __syncthreads();

    if (lane < 16)
        out[(size_t)b * 16 + lane] = red[wave * 16 + lane];
}

extern "C" void kernel_launch(void* const* d_in, const int* in_sizes, int n_in,
                              void* d_out, int out_size, void* d_ws, size_t ws_size,
                              hipStream_t stream) {
    const float* feat1 = (const float*)d_in[0];
    const float* feat2 = (const float*)d_in[1];
    float* out = (float*)d_out;

    dim3 grid(B_TOT / WAVES_PER_BLOCK);     // 512 blocks
    dim3 block(32 * WAVES_PER_BLOCK);       // 256 threads = 8 wave32
    hipLaunchKernelGGL(corr_wmma_f32_kernel, grid, block, 0, stream,
                       feat1, feat2, out);
}